// KNNDist_24790551233435
// MI455X (gfx1250) — compile-verified
//
#include <hip/hip_runtime.h>
#include <math.h>

// WMMA operand vector types (f32 16x16x4: A/B = 2 VGPRs, C/D = 8 VGPRs, wave32)
typedef __attribute__((ext_vector_type(2))) float v2f;
typedef __attribute__((ext_vector_type(4))) float v4f;
typedef __attribute__((ext_vector_type(8))) float v8f;

#define ALPHA_F 1.05f

// Branch-guarded sorted insert of v into ascending 6-list (registers only).
__device__ __forceinline__ void insert6(float v, float& t0, float& t1, float& t2,
                                        float& t3, float& t4, float& t5) {
  if (v < t5) {
    float c5 = fminf(t5, fmaxf(t4, v));
    float c4 = fminf(t4, fmaxf(t3, v));
    float c3 = fminf(t3, fmaxf(t2, v));
    float c2 = fminf(t2, fmaxf(t1, v));
    float c1 = fminf(t1, fmaxf(t0, v));
    float c0 = fminf(t0, v);
    t0 = c0; t1 = c1; t2 = c2; t3 = c3; t4 = c4; t5 = c5;
  }
}

// Pack pc[i] = (x,y,z) into packed[i] = (x, y, z, ||p||^2), 16B stride.
// This makes the WMMA A-operand a single aligned b64 load per lane.
__global__ __launch_bounds__(256) void pack_kernel(const float* __restrict__ pc,
                                                   float* __restrict__ packed,
                                                   int total) {
  const int i = blockIdx.x * 256 + threadIdx.x;
  if (i < total) {
    const float x = pc[3 * i + 0];
    const float y = pc[3 * i + 1];
    const float z = pc[3 * i + 2];
    const float n = fmaf(x, x, fmaf(y, y, z * z));
    v4f o = {x, y, z, n};
    *(v4f*)(packed + 4 * i) = o;   // global_store_b128
  }
}

// One 16x16 tile of squared distances via a single V_WMMA_F32_16X16X4_F32.
// A row m = (x_m, y_m, z_m, ||p_m||^2); B col n = (-2x_n, -2y_n, -2z_n, 1);
// C = ||p_n||^2 broadcast  =>  D[m][n] = ||p_m - p_n||^2.
// A-operand VGPR pair comes straight from one aligned b64 load:
//   lanes 0-15 (K=0,1) read bytes [0,8) = (x,y); lanes 16-31 (K=2,3) read [8,16) = (z,n).
__device__ __forceinline__ v8f dist_tile(const float* __restrict__ ptr,
                                         v2f bmat, float qn) {
  const v2f amat = *(const v2f*)ptr;
  v8f acc = {qn, qn, qn, qn, qn, qn, qn, qn};
  return __builtin_amdgcn_wmma_f32_16x16x4_f32(
      /*neg_a=*/false, amat, /*neg_b=*/false, bmat,
      /*c_mod=*/(short)0, acc, /*reuse_a=*/false, /*reuse_b=*/false);
}

// One wave (32 lanes) per (batch, 16-query tile).
// Computes value[b, i] = mean of 5 nearest-neighbor squared distances (self dropped).
__global__ __launch_bounds__(32) void knn_value_kernel(const float* __restrict__ packed,
                                                       float* __restrict__ value,
                                                       int npts, int ntile) {
  const int itile = blockIdx.x;          // query tile: 16 queries
  const int b     = blockIdx.y;          // batch
  const int lane  = threadIdx.x;         // 0..31 (wave32)
  const int idx   = lane & 15;
  const bool hi   = lane >= 16;
  const unsigned off = hi ? 2u : 0u;     // float offset into 4-float record

  const float* __restrict__ Pp = packed + (size_t)b * (size_t)npts * 4;

  // Query owned by this lane's output column N = idx.
  const int q = itile * 16 + idx;
  const v2f qv = *(const v2f*)(Pp + 4 * q + off);  // lo:(x,y)  hi:(z,n)
  const float qn = Pp[4 * q + 3];                  // ||p_q||^2 for C broadcast

  // B matrix (4x16, K x N): col n = (-2x, -2y, -2z, 1).
  // lo lanes: (-2x,-2y); hi lanes: (-2z, 1). qv.x is x (lo) or z (hi).
  v2f bmat;
  bmat.x = -2.0f * qv.x;
  bmat.y = hi ? 1.0f : (-2.0f * qv.y);

  const float BIG = 3.0e38f;
  float t0 = BIG, t1 = BIG, t2 = BIG, t3 = BIG, t4 = BIG, t5 = BIG;

  // Loop-carried candidate pointer: one 64-bit add per iteration,
  // loads at immediate offsets 0/256/512/768 bytes.
  const float* p = Pp + ((unsigned)idx * 4u + off);

  int jt = 0;
  // Main loop: 4 WMMA tiles (64 candidates) per iteration, one rejection test.
  for (; jt + 4 <= ntile; jt += 4, p += 256) {
    v8f d0 = dist_tile(p +   0, bmat, qn);
    v8f d1 = dist_tile(p +  64, bmat, qn);
    v8f d2 = dist_tile(p + 128, bmat, qn);
    v8f d3 = dist_tile(p + 192, bmat, qn);

    // Branch-free min over the 32 new distances in this lane (v_min3 chains).
    float m[8];
#pragma unroll
    for (int r = 0; r < 8; ++r)
      m[r] = fminf(fminf(d0[r], d1[r]), fminf(d2[r], d3[r]));
    const float mm = fminf(fminf(fminf(m[0], m[1]), fminf(m[2], m[3])),
                           fminf(fminf(m[4], m[5]), fminf(m[6], m[7])));

    // Common case after warm-up: nothing beats the current 6th-smallest -> skip.
    if (mm < t5) {
#pragma unroll
      for (int r = 0; r < 8; ++r) {
        insert6(d0[r], t0, t1, t2, t3, t4, t5);
        insert6(d1[r], t0, t1, t2, t3, t4, t5);
        insert6(d2[r], t0, t1, t2, t3, t4, t5);
        insert6(d3[r], t0, t1, t2, t3, t4, t5);
      }
    }
  }
  // Tail (not taken for N = 4096).
  for (; jt < ntile; ++jt, p += 64) {
    v8f d = dist_tile(p, bmat, qn);
#pragma unroll
    for (int r = 0; r < 8; ++r) insert6(d[r], t0, t1, t2, t3, t4, t5);
  }

  // Merge with partner half-lane (same query, other 8 candidate rows per tile).
  // Snapshot partner values BEFORE mutating our own list.
  float u0 = __shfl_xor(t0, 16, 32);
  float u1 = __shfl_xor(t1, 16, 32);
  float u2 = __shfl_xor(t2, 16, 32);
  float u3 = __shfl_xor(t3, 16, 32);
  float u4 = __shfl_xor(t4, 16, 32);
  float u5 = __shfl_xor(t5, 16, 32);
  insert6(u0, t0, t1, t2, t3, t4, t5);
  insert6(u1, t0, t1, t2, t3, t4, t5);
  insert6(u2, t0, t1, t2, t3, t4, t5);
  insert6(u3, t0, t1, t2, t3, t4, t5);
  insert6(u4, t0, t1, t2, t3, t4, t5);
  insert6(u5, t0, t1, t2, t3, t4, t5);

  // Drop smallest (self distance ~0), mean of remaining 5.
  const float mean5 = (t1 + t2 + t3 + t4 + t5) * 0.2f;
  if (!hi) value[(size_t)b * npts + (size_t)itile * 16 + idx] = mean5;
}

// Single block: per-batch mean/std(ddof=1) threshold, masked mean, weight, average.
__global__ __launch_bounds__(256) void knn_reduce_kernel(const float* __restrict__ value,
                                                         const float* __restrict__ weights,
                                                         float* __restrict__ out,
                                                         int nbatch, int npts) {
  __shared__ float sA[256];
  __shared__ float sB[256];
  const int tid = threadIdx.x;

  float total = 0.0f;
  for (int b = 0; b < nbatch; ++b) {
    const float* v = value + (size_t)b * npts;

    float s = 0.0f, s2 = 0.0f;
    for (int i = tid; i < npts; i += 256) {
      float x = v[i];
      s += x;
      s2 = fmaf(x, x, s2);
    }
    sA[tid] = s; sB[tid] = s2;
    __syncthreads();
    for (int off = 128; off > 0; off >>= 1) {
      if (tid < off) { sA[tid] += sA[tid + off]; sB[tid] += sB[tid + off]; }
      __syncthreads();
    }
    const float S = sA[0];
    const float S2 = sB[0];
    __syncthreads();

    const float n    = (float)npts;
    const float mean = S / n;
    const float var  = (S2 - n * mean * mean) / (n - 1.0f);
    const float thr  = mean + ALPHA_F * sqrtf(fmaxf(var, 0.0f));

    float ms = 0.0f;
    for (int i = tid; i < npts; i += 256) {
      float x = v[i];
      ms += (x > thr) ? x : 0.0f;
    }
    sA[tid] = ms;
    __syncthreads();
    for (int off = 128; off > 0; off >>= 1) {
      if (tid < off) sA[tid] += sA[tid + off];
      __syncthreads();
    }
    total += (sA[0] / n) * weights[b];
    __syncthreads();
  }

  if (tid == 0) out[0] = total / (float)nbatch;
}

extern "C" void kernel_launch(void* const* d_in, const int* in_sizes, int n_in,
                              void* d_out, int out_size, void* d_ws, size_t ws_size,
                              hipStream_t stream) {
  const float* pc      = (const float*)d_in[0];   // [B, N, 3] f32
  const float* weights = (const float*)d_in[1];   // [B] f32
  float* out = (float*)d_out;                     // scalar f32

  const int B = in_sizes[1];                      // 8
  const int N = in_sizes[0] / (B * 3);            // 4096
  const int NT = N / 16;                          // 256 tiles
  const int total = B * N;

  // Workspace layout: packed [B*N*4] floats, then value [B*N] floats.
  float* packed = (float*)d_ws;
  float* value  = packed + (size_t)total * 4;

  pack_kernel<<<(total + 255) / 256, 256, 0, stream>>>(pc, packed, total);

  dim3 grid(NT, B);
  knn_value_kernel<<<grid, 32, 0, stream>>>(packed, value, N, NT);
  knn_reduce_kernel<<<1, 256, 0, stream>>>(value, weights, out, B, N);
}